// DiscriminatorForMissing_53291954208814
// MI455X (gfx1250) — compile-verified
//
#include <hip/hip_runtime.h>
#include <hip/hip_bf16.h>
#include <math.h>

// Problem sizes (fixed by the reference)
#define B_SZ  8192
#define D_SZ  512
#define H0_SZ 1024
#define H1_SZ 1024
#define K_SZ  10

#define INV_SQRT_2PI 0.3989422804014327f
#define INV_SQRT_2   0.70710678118654752f
#define LOG_2PI_F    1.8378770664093453f

typedef __attribute__((ext_vector_type(2))) float v2f;
typedef __attribute__((ext_vector_type(8))) float v8f;

// V_WMMA_F32_16X16X4_F32: D(16x16 f32, 8 VGPR) = A(16x4 f32, 2 VGPR) * B(4x16 f32, 2 VGPR) + C
__device__ __forceinline__ v8f wmma_f32(v2f a, v2f b, v8f c) {
  return __builtin_amdgcn_wmma_f32_16x16x4_f32(false, a, false, b, (short)0, c,
                                               false, false);
}

// ---------------------------------------------------------------------------
// Kernel 0: init out[b] = b3, zero the column-stats scratch
// ---------------------------------------------------------------------------
__global__ void k_init(float* __restrict__ out, const float* __restrict__ b3,
                       float* __restrict__ stats) {
  int i = blockIdx.x * blockDim.x + threadIdx.x;
  if (i < B_SZ) out[i] = b3[0];
  if (i < 3 * D_SZ) stats[i] = 0.f;
}

// ---------------------------------------------------------------------------
// Kernel 1: NaN-masked per-column moments of x: cnt_d, S1_d, S2_d
// block = 512 threads (one per column, coalesced), grid = B/64 row chunks
// ---------------------------------------------------------------------------
__global__ void k_stats(const float* __restrict__ x, float* __restrict__ stats) {
  const int d = threadIdx.x;
  const int r0 = blockIdx.x * 64;
  float cnt = 0.f, s1 = 0.f, s2 = 0.f;
  for (int i = 0; i < 64; ++i) {
    float v = x[(size_t)(r0 + i) * D_SZ + d];
    if (v == v) { cnt += 1.f; s1 += v; s2 += v * v; }
  }
  atomicAdd(&stats[d], cnt);
  atomicAdd(&stats[D_SZ + d], s1);
  atomicAdd(&stats[2 * D_SZ + d], s2);
}

// ---------------------------------------------------------------------------
// Kernel 2: q[k] from column stats, then r = softmax(-0.5 q + log_p). Tiny.
// ---------------------------------------------------------------------------
__global__ void k_q(const float* __restrict__ stats,
                    const float* __restrict__ means,
                    const float* __restrict__ covs,
                    const float* __restrict__ wmix,
                    const float* __restrict__ gamma_p,
                    float* __restrict__ r) {
  __shared__ float red[D_SZ];
  __shared__ float qsh[K_SZ];
  __shared__ float totalCnt;
  const int d = threadIdx.x;
  const float gamma = gamma_p[0];
  const float cnt = stats[d];
  const float s1 = stats[D_SZ + d];
  const float s2 = stats[2 * D_SZ + d];

  red[d] = cnt;
  __syncthreads();
  for (int s = D_SZ / 2; s > 0; s >>= 1) {
    if (d < s) red[d] += red[d + s];
    __syncthreads();
  }
  if (d == 0) totalCnt = red[0];
  __syncthreads();

  for (int k = 0; k < K_SZ; ++k) {
    float m = means[k * D_SZ + d];
    float c = covs[k * D_SZ + d];
    float qf = gamma + c;  // qfull at non-NaN entries; NaN entries contribute 0
    float term = cnt * logf(qf) + (s2 - 2.f * m * s1 + cnt * m * m) / qf;
    red[d] = term;
    __syncthreads();
    for (int s = D_SZ / 2; s > 0; s >>= 1) {
      if (d < s) red[d] += red[d + s];
      __syncthreads();
    }
    if (d == 0) qsh[k] = red[0] + totalCnt * LOG_2PI_F;
    __syncthreads();
  }

  if (d == 0) {
    float mx = -INFINITY;
    for (int k = 0; k < K_SZ; ++k) mx = fmaxf(mx, wmix[k]);
    float se = 0.f;
    for (int k = 0; k < K_SZ; ++k) se += expf(wmix[k] - mx);
    float lse = mx + logf(se);
    float lq[K_SZ];
    float mq = -INFINITY;
    for (int k = 0; k < K_SZ; ++k) {
      lq[k] = -0.5f * qsh[k] + (wmix[k] - lse);
      mq = fmaxf(mq, lq[k]);
    }
    float ss = 0.f;
    for (int k = 0; k < K_SZ; ++k) { lq[k] = expf(lq[k] - mq); ss += lq[k]; }
    for (int k = 0; k < K_SZ; ++k) r[k] = lq[k] / ss;
  }
}

// ---------------------------------------------------------------------------
// Kernel 3: fused layer1/sigma/nr/mixture kernel.
// One 16x16 (b,h0) tile per wave; 4 waves/block cover 64 h0 columns.
// 21 WMMA accumulators: base = xz*W1^T, and per k: mk*(means_k . W1)^T,
// mk*(covs_k . W1sq)^T. Epilogue applies relu, nr(), and sum_k r[k]*distrib.
// ---------------------------------------------------------------------------
__global__ __launch_bounds__(128, 1) void k_fused(
    const float* __restrict__ x, const float* __restrict__ means,
    const float* __restrict__ covs, const float* __restrict__ W1,
    const float* __restrict__ b1, const float* __restrict__ r,
    float* __restrict__ h) {
  __shared__ __align__(16) float sM[K_SZ * D_SZ];
  __shared__ __align__(16) float sC[K_SZ * D_SZ];
  for (int i = threadIdx.x; i < K_SZ * D_SZ; i += 128) {
    sM[i] = means[i];
    sC[i] = covs[i];
  }
  __syncthreads();

  const int lane = threadIdx.x & 31;
  const int wave = threadIdx.x >> 5;
  const int l16 = lane & 15;
  const int lh = lane >> 4;
  const int b0 = blockIdx.x * 16;
  const int h0 = (blockIdx.y * 4 + wave) * 16;

  v8f accB = {};
  v8f accM[K_SZ];
  v8f accS[K_SZ];
#pragma unroll
  for (int k = 0; k < K_SZ; ++k) {
    accM[k] = (v8f){};
    accS[k] = (v8f){};
  }

  const float* xrow = x + (size_t)(b0 + l16) * D_SZ;
  const float* wrow = W1 + (size_t)(h0 + l16) * D_SZ;

  for (int d0 = 0; d0 < D_SZ; d0 += 4) {
    const int da = d0 + 2 * lh;  // this lane's K-rows (ISA A/B fragment layout)

    // A operands: value fragment (NaN->0) and NaN-mask fragment
    float2 xv = *(const float2*)(xrow + da);
    const bool n0 = xv.x != xv.x, n1 = xv.y != xv.y;
    v2f xz = {n0 ? 0.f : xv.x, n1 ? 0.f : xv.y};
    v2f mk = {n0 ? 1.f : 0.f, n1 ? 1.f : 0.f};

    // B operand: W1 fragment (B[kd][n] = W1[h0+n][kd])
    float2 wv = *(const float2*)(wrow + da);
    v2f w1 = {wv.x, wv.y};
    v2f w1sq = {wv.x * wv.x, wv.y * wv.y};

    accB = wmma_f32(xz, w1, accB);

#pragma unroll
    for (int k = 0; k < K_SZ; ++k) {
      float2 mv = *(const float2*)(&sM[k * D_SZ + da]);
      float2 cv = *(const float2*)(&sC[k * D_SZ + da]);
      v2f bm = {w1.x * mv.x, w1.y * mv.y};
      v2f bc = {w1sq.x * cv.x, w1sq.y * cv.y};
      accM[k] = wmma_f32(mk, bm, accM[k]);
      accS[k] = wmma_f32(mk, bc, accS[k]);
    }
  }

  // Epilogue: relu-moment + mixture weighting, write h[b, h0].
  // sigma >= 0 by construction, so s=sqrt(sg) and w=mu/s are both derived
  // from a single v_rsq_f32 (no IEEE sqrt-refine / divide sequences).
  const float b1v = b1[h0 + l16];
  float rk[K_SZ];
#pragma unroll
  for (int k = 0; k < K_SZ; ++k) rk[k] = r[k];

#pragma unroll
  for (int j = 0; j < 8; ++j) {
    const int brow = b0 + j + 8 * lh;  // C layout: VGPR j -> M=j(+8), lane%16 -> N
    float acc = 0.f;
#pragma unroll
    for (int k = 0; k < K_SZ; ++k) {
      float mu = fmaxf(accB[j] + accM[k][j] + b1v, 0.f);
      float sg = accS[k][j];
      float val = mu;  // sigma==0 fallback: 0.5*(mu+|mu|) = mu since mu >= 0
      if (sg != 0.f) {
        float inv = __builtin_amdgcn_rsqf(sg);  // v_rsq_f32
        float s = sg * inv;                     // sqrt(sg)
        float w = mu * inv;                     // mu / sqrt(sg)
        val = s * (__expf(-0.5f * w * w) * INV_SQRT_2PI +
                   0.5f * w * (1.f + erff(w * INV_SQRT_2)));
      }
      acc = fmaf(rk[k], val, acc);
    }
    h[(size_t)brow * H0_SZ + h0 + l16] = acc;
  }
}

// ---------------------------------------------------------------------------
// Kernel 4: h2 = leaky_relu(h @ W2^T + b2, 0.2); out += h2 @ W3^T
// One 16x16 (b,h1) tile per wave; fused W3 dot via shfl reduction + atomics.
// ---------------------------------------------------------------------------
__global__ __launch_bounds__(128, 1) void k_mlp(
    const float* __restrict__ h, const float* __restrict__ W2,
    const float* __restrict__ b2, const float* __restrict__ W3,
    float* __restrict__ out) {
  const int lane = threadIdx.x & 31;
  const int wave = threadIdx.x >> 5;
  const int l16 = lane & 15;
  const int lh = lane >> 4;
  const int b0 = blockIdx.x * 16;
  const int n0 = (blockIdx.y * 4 + wave) * 16;

  v8f acc = {};
  const float* hrow = h + (size_t)(b0 + l16) * H0_SZ;
  const float* wrow = W2 + (size_t)(n0 + l16) * H0_SZ;
  for (int kk = 0; kk < H0_SZ; kk += 4) {
    const int ka = kk + 2 * lh;
    float2 av = *(const float2*)(hrow + ka);
    float2 bv = *(const float2*)(wrow + ka);
    v2f a = {av.x, av.y};
    v2f b = {bv.x, bv.y};
    acc = wmma_f32(a, b, acc);
  }

  const float b2v = b2[n0 + l16];
  const float w3v = W3[n0 + l16];
#pragma unroll
  for (int j = 0; j < 8; ++j) {
    float v = acc[j] + b2v;
    v = (v > 0.f) ? v : 0.2f * v;
    float p = v * w3v;
#pragma unroll
    for (int off = 1; off < 16; off <<= 1) p += __shfl_xor(p, off, 16);
    if (l16 == 0) atomicAdd(&out[b0 + j + 8 * lh], p);
  }
}

// ---------------------------------------------------------------------------
extern "C" void kernel_launch(void* const* d_in, const int* in_sizes, int n_in,
                              void* d_out, int out_size, void* d_ws,
                              size_t ws_size, hipStream_t stream) {
  const float* x     = (const float*)d_in[0];
  const float* means = (const float*)d_in[1];
  const float* covs  = (const float*)d_in[2];
  const float* wmix  = (const float*)d_in[3];
  const float* gam   = (const float*)d_in[4];
  const float* W1    = (const float*)d_in[5];
  const float* b1    = (const float*)d_in[6];
  const float* W2    = (const float*)d_in[7];
  const float* b2    = (const float*)d_in[8];
  const float* W3    = (const float*)d_in[9];
  const float* b3    = (const float*)d_in[10];
  float* out = (float*)d_out;

  // Workspace layout: [stats 3*D][r K] (first 8KB), then h [B*H0] (32MB)
  float* stats = (float*)d_ws;
  float* r     = stats + 3 * D_SZ;
  float* h     = (float*)((char*)d_ws + 8192);

  k_init<<<(B_SZ + 255) / 256, 256, 0, stream>>>(out, b3, stats);
  k_stats<<<B_SZ / 64, D_SZ, 0, stream>>>(x, stats);
  k_q<<<1, D_SZ, 0, stream>>>(stats, means, covs, wmix, gam, r);

  dim3 g3(B_SZ / 16, H0_SZ / 64);
  k_fused<<<g3, 128, 0, stream>>>(x, means, covs, W1, b1, r, h);

  dim3 g4(B_SZ / 16, H1_SZ / 64);
  k_mlp<<<g4, 128, 0, stream>>>(h, W2, b2, W3, out);
}